// PointnetFPModule2_19069654794726
// MI455X (gfx1250) — compile-verified
//
#include <hip/hip_runtime.h>
#include <hip/hip_bf16.h>

// Problem constants (match reference)
#define BATCH 8
#define NQ    8192
#define MK    2048
#define CH    256
#define EPSF  1e-8f

typedef float v2f __attribute__((ext_vector_type(2)));
typedef float v8f __attribute__((ext_vector_type(8)));

// Branchy-but-rare top-3 insert (sorted t0 <= t1 <= t2). Outer test compiles
// to a v_cmpx + s_cbranch_execz exec-mask branch.
#define TOP3_INSERT(sv, id)                                        \
  do {                                                             \
    float _s = (sv); int _i = (id);                                \
    if (_s < t2) {                                                 \
      if (_s < t1) {                                               \
        t2 = t1; i2 = i1;                                          \
        if (_s < t0) { t1 = t0; i1 = i0; t0 = _s; i0 = _i; }       \
        else         { t1 = _s; i1 = _i; }                         \
      } else { t2 = _s; i2 = _i; }                                 \
    }                                                              \
  } while (0)

// ---------------------------------------------------------------------------
// Stage 1: three_nn via V_WMMA_F32_16X16X4_F32.
// Score = ||k||^2 - 2 u.k computed entirely by the matrix pipe:
//   A row (known m)  = (-2kx, -2ky, -2kz, ||k||^2)   [16x4, K-halves split
//                      across lane halves per ISA layout]
//   B col (query n)  = ( ux ,  uy ,  uz ,   1   )    [4x16]
// D[m][n] = score; ||u||^2 is per-query constant so ranking is unaffected and
// it is added back only for the 3 winners' weights.
// One wave = 16 queries; 2x-unrolled sweep over 128 known tiles of 16.
// Block = 256 threads (8 waves, 128 queries) to amortize the LDS staging.
// ---------------------------------------------------------------------------
__global__ __launch_bounds__(256) void three_nn_wmma_kernel(
    const float* __restrict__ unknown,   // (B, NQ, 3)
    const float* __restrict__ known,     // (B, MK, 3)
    float* __restrict__ w_ws,            // (B, NQ, 3) normalized weights
    int*   __restrict__ idx_ws)          // (B, NQ, 3) neighbor indices
{
  __shared__ float4 s_kn[MK];  // (-2kx, -2ky, -2kz, |k|^2), 32 KB

  const int tilesPerBatch = NQ / 128;             // 128 queries per block
  const int b = blockIdx.x / tilesPerBatch;
  const int t = blockIdx.x % tilesPerBatch;

  // Cooperative stage of all knowns for this batch into LDS, pre-transformed.
  const float* kb = known + (size_t)b * MK * 3;
  for (int m = threadIdx.x; m < MK; m += blockDim.x) {
    float kx = kb[m * 3 + 0];
    float ky = kb[m * 3 + 1];
    float kz = kb[m * 3 + 2];
    s_kn[m] = make_float4(-2.0f * kx, -2.0f * ky, -2.0f * kz,
                          kx * kx + ky * ky + kz * kz);
  }
  __syncthreads();

  const int lane = threadIdx.x & 31;
  const int wv   = threadIdx.x >> 5;
  const int half = lane >> 4;                     // K-half (ISA A/B layout)
  const int q    = t * 128 + wv * 16 + (lane & 15);

  // Full query coords per lane (also needed for ||u||^2).
  const float* up = unknown + ((size_t)b * NQ + q) * 3;
  float ux = up[0], uy = up[1], uz = up[2];
  float un = ux * ux + uy * uy + uz * uz;

  // B fragment (4x16 f32): lanes 0-15 -> (K0,K1)=(ux,uy); 16-31 -> (K2,K3)=(uz,1)
  v2f bfrag;
  bfrag.x = half ? uz : ux;
  bfrag.y = half ? 1.0f : uy;

  float t0 = 3.4e38f, t1 = 3.4e38f, t2 = 3.4e38f;
  int   i0 = 0, i1 = 0, i2 = 0;

  const float* s_knf = (const float*)s_kn;
  const int aoff = (lane & 15) * 4 + half * 2;    // floats into float4 table

  for (int m0 = 0; m0 < MK; m0 += 32) {
    // Two independent A fragments -> both ds_loads in flight, two WMMAs
    // back-to-back; insert VALU of tile 0 co-executes with WMMA of tile 1.
    v2f a0 = *(const v2f*)(s_knf + m0 * 4 + aoff);
    v2f a1 = *(const v2f*)(s_knf + (m0 + 16) * 4 + aoff);
    v8f cz0 = {};
    v8f cz1 = {};
    v8f d0 = __builtin_amdgcn_wmma_f32_16x16x4_f32(
        false, a0, false, bfrag, (short)0, cz0, false, false);
    v8f d1 = __builtin_amdgcn_wmma_f32_16x16x4_f32(
        false, a1, false, bfrag, (short)0, cz1, false, false);

    // D layout: lanes<16 hold knowns m0+0..7; lanes>=16 hold m0+8..15.
    const int mb0 = m0 + half * 8;
    const int mb1 = mb0 + 16;
#pragma unroll
    for (int v = 0; v < 8; ++v) {
      float sv = d0[v];
      TOP3_INSERT(sv, mb0 + v);
    }
#pragma unroll
    for (int v = 0; v < 8; ++v) {
      float sv = d1[v];
      TOP3_INSERT(sv, mb1 + v);
    }
  }

  // Merge lane L with lane L^16 (same query, other half of the knowns).
  {
    float p; int pi;
    p = __shfl_xor(t0, 16, 32); pi = __shfl_xor(i0, 16, 32); TOP3_INSERT(p, pi);
    p = __shfl_xor(t1, 16, 32); pi = __shfl_xor(i1, 16, 32); TOP3_INSERT(p, pi);
    p = __shfl_xor(t2, 16, 32); pi = __shfl_xor(i2, 16, 32); TOP3_INSERT(p, pi);
  }

  if (lane < 16) {
    // Recover true squared distances and form normalized inverse-d2 weights.
    float r0 = 1.0f / (t0 + un + EPSF);
    float r1 = 1.0f / (t1 + un + EPSF);
    float r2 = 1.0f / (t2 + un + EPSF);
    float inv = 1.0f / (r0 + r1 + r2);
    size_t base = ((size_t)b * NQ + q) * 3;
    w_ws[base + 0] = r0 * inv;
    w_ws[base + 1] = r1 * inv;
    w_ws[base + 2] = r2 * inv;
    idx_ws[base + 0] = i0;
    idx_ws[base + 1] = i1;
    idx_ws[base + 2] = i2;
  }
}

// ---------------------------------------------------------------------------
// Stage 2: three_interpolate. out[b,c,n] = sum_k w_k * feats[b,c,idx_k].
// Each thread owns 4 consecutive n -> global_store_b128 on the 67 MB output
// stream (the dominant traffic). Gathers hit an 8 KB per-channel feature row
// resident in near caches / L2 (whole feats tensor << 192 MB L2).
// Channel range split over gridDim.z for occupancy.
// ---------------------------------------------------------------------------
__global__ __launch_bounds__(256) void three_interp_kernel(
    const float* __restrict__ feats,   // (B, CH, MK)
    const float* __restrict__ w_ws,    // (B, NQ, 3)
    const int*   __restrict__ idx_ws,  // (B, NQ, 3)
    float* __restrict__ out)           // (B, CH, NQ)
{
  const int n0 = (blockIdx.x * blockDim.x + threadIdx.x) * 4;
  const int b  = blockIdx.y;
  const int c0 = blockIdx.z * (CH / 4);

  float w0[4], w1[4], w2[4];
  int   j0[4], j1[4], j2[4];
#pragma unroll
  for (int j = 0; j < 4; ++j) {
    size_t base = ((size_t)b * NQ + n0 + j) * 3;
    w0[j] = w_ws[base + 0]; w1[j] = w_ws[base + 1]; w2[j] = w_ws[base + 2];
    j0[j] = idx_ws[base + 0]; j1[j] = idx_ws[base + 1]; j2[j] = idx_ws[base + 2];
  }

  const float* F = feats + (size_t)b * CH * MK;
  float*       O = out   + (size_t)b * CH * NQ;

  for (int c = c0; c < c0 + CH / 4; ++c) {
    const float* Fc = F + (size_t)c * MK;
    float4 o;
    o.x = w0[0] * Fc[j0[0]] + w1[0] * Fc[j1[0]] + w2[0] * Fc[j2[0]];
    o.y = w0[1] * Fc[j0[1]] + w1[1] * Fc[j1[1]] + w2[1] * Fc[j2[1]];
    o.z = w0[2] * Fc[j0[2]] + w1[2] * Fc[j1[2]] + w2[2] * Fc[j2[2]];
    o.w = w0[3] * Fc[j0[3]] + w1[3] * Fc[j1[3]] + w2[3] * Fc[j2[3]];
    *(float4*)(O + (size_t)c * NQ + n0) = o;   // global_store_b128
  }
}

// ---------------------------------------------------------------------------
extern "C" void kernel_launch(void* const* d_in, const int* in_sizes, int n_in,
                              void* d_out, int out_size, void* d_ws, size_t ws_size,
                              hipStream_t stream) {
  const float* unknown = (const float*)d_in[0];  // (B, NQ, 3)
  const float* known   = (const float*)d_in[1];  // (B, MK, 3)
  const float* feats   = (const float*)d_in[2];  // (B, CH, MK)
  float* out = (float*)d_out;                    // (B, CH, NQ)

  float* w_ws   = (float*)d_ws;                                           // 786 KB
  int*   idx_ws = (int*)((char*)d_ws + (size_t)BATCH * NQ * 3 * sizeof(float));

  dim3 g1(BATCH * (NQ / 128));
  three_nn_wmma_kernel<<<g1, 256, 0, stream>>>(unknown, known, w_ws, idx_ws);

  dim3 g2(NQ / (256 * 4), BATCH, 4);
  three_interp_kernel<<<g2, 256, 0, stream>>>(feats, w_ws, idx_ws, out);
}